// Interpolate1D_87969520157065
// MI455X (gfx1250) — compile-verified
//
#include <hip/hip_runtime.h>

// ---------------------------------------------------------------------------
// Stride-2 linear 1D upsample, (16, 8192, 256) f32 -> (16, 16384, 256) f32.
// Pure bandwidth problem (AI ~ 0.17 flop/byte): 384 MiB of traffic -> ~16.5us
// floor at 23.3 TB/s. Data path:
//   - TDM tensor_load_to_lds DMAs a 33-row x 1KB tile into LDS (single issue
//     by wave 0; TENSORcnt + workgroup barrier publishes it).
//   - Each thread owns one float4 column over 8 CONSECUTIVE rows, so each
//     iteration needs only one new ds_load_b128 (running-register reuse).
//   - Outputs leave via b128 non-temporal stores (256 MiB stream > 192MB L2).
// ---------------------------------------------------------------------------

typedef float        v4f   __attribute__((ext_vector_type(4)));
typedef unsigned int u32x4 __attribute__((ext_vector_type(4)));
typedef int          i32x8 __attribute__((ext_vector_type(8)));
typedef int          i32x4 __attribute__((ext_vector_type(4)));

#define TILE_ROWS   32
#define ROWS_PER_T  8             // consecutive rows per thread
#define ROW_FLOATS  256
#define ROW_V4      64            // 256 floats = 64 float4
#define N_PER_BATCH 8192

__global__ __launch_bounds__(256) void Interpolate1D_kernel(
    const float* __restrict__ in, float* __restrict__ out,
    unsigned long long total_rows) {

  __shared__ v4f lds4[(TILE_ROWS + 1) * ROW_V4];   // 33 KB tile + halo row

  const unsigned long long n0 = (unsigned long long)blockIdx.x * TILE_ROWS;

  // ---- Producer: wave 0 issues one TDM load of rows [n0, n0+33) ----------
  if (threadIdx.x < 32) {
    unsigned long long gaddr =
        (unsigned long long)(size_t)in + n0 * (ROW_FLOATS * 4ull);
    // Low 32 bits of a generic pointer to LDS are the wave-relative LDS byte
    // address (ISA 10.2 aperture mapping).
    unsigned int lds_addr = (unsigned int)(size_t)(&lds4[0]);
    unsigned int tdim1    = (unsigned int)(total_rows - n0); // OOB rows -> 0

    // D# group 0 (ISA 8.3): count=1 | lds_addr | global_addr[56:0] | type=2
    u32x4 g0;
    g0[0] = 1u;
    g0[1] = lds_addr;
    g0[2] = (unsigned int)(gaddr & 0xFFFFFFFFull);
    g0[3] = (unsigned int)((gaddr >> 32) & 0x01FFFFFFull) | (2u << 30);

    // D# group 1 (ISA 8.4): data_size=4B, tensor_dim0=256, tensor_dim1=tdim1,
    // tile_dim0=256, tile_dim1=33, tensor_dim0_stride=256.
    i32x8 g1;
    g1[0] = (int)(2u << 16);                                   // data_size=2 (4B)
    g1[1] = (int)(((unsigned)ROW_FLOATS & 0xFFFFu) << 16);     // tensor_dim0 lo
    g1[2] = (int)((tdim1 & 0xFFFFu) << 16);                    // tensor_dim1 lo
    g1[3] = (int)(((tdim1 >> 16) & 0xFFFFu)                    // tensor_dim1 hi
                  | (((unsigned)ROW_FLOATS & 0xFFFFu) << 16)); // tile_dim0
    g1[4] = (int)(TILE_ROWS + 1);                              // tile_dim1=33
    g1[5] = ROW_FLOATS;                                        // dim0_stride lo
    g1[6] = 0;
    g1[7] = 0;

    i32x4 g2 = {0, 0, 0, 0};   // 2D tensor: groups 2/3 unused
    i32x4 g3 = {0, 0, 0, 0};
    i32x8 g4 = {0, 0, 0, 0, 0, 0, 0, 0};  // 6-arg toolchain extra group, unused

    __builtin_amdgcn_tensor_load_to_lds(g0, g1, g2, g3, g4, /*cpol=*/0);
    __builtin_amdgcn_s_wait_tensorcnt(0);
  }
  __syncthreads();

  // ---- Consumer: one float4 column, 8 consecutive rows, running reuse ----
  const int c4    = threadIdx.x & (ROW_V4 - 1);
  const int nbase = (threadIdx.x >> 6) * ROWS_PER_T;   // 0, 8, 16, 24

  // out flat float4 index for (row 2r, col c4) = (n0 + n)*2*ROW_V4 + c4
  v4f* __restrict__ ob = (v4f*)out + n0 * (2 * ROW_V4) + (unsigned)c4;

  const v4f* __restrict__ lrow = &lds4[nbase * ROW_V4 + c4];
  v4f cur = *lrow;                                     // row nbase

  // Batch boundary (odd sample = x[N-1]) can only occur at local row 31.
  const bool tile_ends_batch =
      (((n0 + (TILE_ROWS - 1)) & (N_PER_BATCH - 1)) == (N_PER_BATCH - 1));

#pragma unroll
  for (int i = 0; i < ROWS_PER_T; ++i) {
    const int n = nbase + i;
    v4f nxt = lrow[(i + 1) * ROW_V4];                  // row n+1 (halo for n=31)
    v4f mid = 0.5f * (cur + nxt);
    if (tile_ends_batch && n == TILE_ROWS - 1) mid = cur;  // out[2N-1] = x[N-1]
    __builtin_nontemporal_store(cur, ob + (unsigned)(n * (2 * ROW_V4)));
    __builtin_nontemporal_store(mid, ob + (unsigned)(n * (2 * ROW_V4) + ROW_V4));
    cur = nxt;
  }
}

extern "C" void kernel_launch(void* const* d_in, const int* in_sizes, int n_in,
                              void* d_out, int out_size, void* d_ws, size_t ws_size,
                              hipStream_t stream) {
  const float* in  = (const float*)d_in[0];
  float*       out = (float*)d_out;
  unsigned long long total_rows =
      (unsigned long long)in_sizes[0] / ROW_FLOATS;          // 16*8192 = 131072
  unsigned int blocks =
      (unsigned int)((total_rows + TILE_ROWS - 1) / TILE_ROWS); // 4096
  Interpolate1D_kernel<<<blocks, 256, 0, stream>>>(in, out, total_rows);
}